// MetaViterbiNetDetector_14508399525921
// MI455X (gfx1250) — compile-verified
//
#include <hip/hip_runtime.h>

#define HIDDEN 128
#define NSTATE 64
#define CHUNK  1024
#define MPINF  1e30f

typedef __attribute__((ext_vector_type(2))) float v2f;
typedef __attribute__((ext_vector_type(8))) float v8f;

// ---------------------------------------------------------------------------
// Lexicographic (value, index) min-reduction across a wave32.
// Matches jnp.argmin tie-breaking (lowest index wins on equal values).
// ---------------------------------------------------------------------------
static __device__ __forceinline__ int wave_argmin32(float v, int lane) {
  float mval = v;
  int   midx = lane;
#pragma unroll
  for (int off = 16; off > 0; off >>= 1) {
    float ov = __shfl_xor(mval, off, 32);
    int   oi = __shfl_xor(midx, off, 32);
    if (ov < mval || (ov == mval && oi < midx)) { mval = ov; midx = oi; }
  }
  return midx;
}

// ---------------------------------------------------------------------------
// Kernel 1: llr[t][s] = -(b2[s] + sum_h relu(rx[t]*w1[h]+b1[h]) * w2[s][h])
// One wave handles a 16-timestep tile, K=128 swept in 32 steps of
// v_wmma_f32_16x16x4_f32, 4 accumulators cover the 64 states.
// A 16x4 f32 layout: lane L holds M=L&15, VGPR v holds K = v + 2*(L>=16).
// B  4x16 f32 layout: lane L holds N=L&15, VGPR v holds K = v + 2*(L>=16).
// C/D 16x16 f32     : VGPR r holds M = r + 8*(L>=16), N = L&15.
// ---------------------------------------------------------------------------
__global__ __launch_bounds__(256) void mlp_llr_wmma(
    const float* __restrict__ rx, const float* __restrict__ w1,
    const float* __restrict__ b1, const float* __restrict__ w2,
    const float* __restrict__ b2, float* __restrict__ llr, int T) {
  const int wave = blockIdx.x * (blockDim.x >> 5) + (threadIdx.x >> 5);
  const int lane = threadIdx.x & 31;
  const int t0   = wave << 4;
  if (t0 >= T) return;                 // wave-uniform: EXEC stays all-ones

  const int   m    = lane & 15;
  const int   half = lane >> 4;
  const float x    = rx[t0 + m];

  const float2* w1v = (const float2*)w1;
  const float2* b1v = (const float2*)b1;
  const float2* w2v = (const float2*)w2;

  v8f acc0 = {0.f,0.f,0.f,0.f,0.f,0.f,0.f,0.f};
  v8f acc1 = {0.f,0.f,0.f,0.f,0.f,0.f,0.f,0.f};
  v8f acc2 = {0.f,0.f,0.f,0.f,0.f,0.f,0.f,0.f};
  v8f acc3 = {0.f,0.f,0.f,0.f,0.f,0.f,0.f,0.f};

#pragma unroll 4
  for (int kk = 0; kk < 32; ++kk) {
    const int hp = kk * 2 + half;      // float2 index: h0 = 4*kk + 2*half
    float2 w1p = w1v[hp];
    float2 b1p = b1v[hp];
    v2f A;
    A[0] = fmaxf(fmaf(x, w1p.x, b1p.x), 0.0f);
    A[1] = fmaxf(fmaf(x, w1p.y, b1p.y), 0.0f);

    float2 q;
    v2f B0, B1, B2, B3;
    q = w2v[(size_t)(m +  0) * 64 + hp]; B0[0] = q.x; B0[1] = q.y;
    q = w2v[(size_t)(m + 16) * 64 + hp]; B1[0] = q.x; B1[1] = q.y;
    q = w2v[(size_t)(m + 32) * 64 + hp]; B2[0] = q.x; B2[1] = q.y;
    q = w2v[(size_t)(m + 48) * 64 + hp]; B3[0] = q.x; B3[1] = q.y;

    acc0 = __builtin_amdgcn_wmma_f32_16x16x4_f32(false, A, false, B0, (short)0, acc0, false, false);
    acc1 = __builtin_amdgcn_wmma_f32_16x16x4_f32(false, A, false, B1, (short)0, acc1, false, false);
    acc2 = __builtin_amdgcn_wmma_f32_16x16x4_f32(false, A, false, B2, (short)0, acc2, false, false);
    acc3 = __builtin_amdgcn_wmma_f32_16x16x4_f32(false, A, false, B3, (short)0, acc3, false, false);
  }

  const float bb0 = b2[m], bb1 = b2[m + 16], bb2 = b2[m + 32], bb3 = b2[m + 48];
#pragma unroll
  for (int r = 0; r < 8; ++r) {
    size_t t   = (size_t)t0 + (size_t)r + (size_t)half * 8;
    float* row = llr + t * NSTATE + m;
    row[0]  = -(acc0[r] + bb0);
    row[16] = -(acc1[r] + bb1);
    row[32] = -(acc2[r] + bb2);
    row[48] = -(acc3[r] + bb3);
  }
}

// ---------------------------------------------------------------------------
// Phase 1: per-chunk 32x32 min-plus transition matrix.
// Lane l propagates column l (start state l) fully in registers; llr pairs
// broadcast with lane shuffles. P[c][s][l] = best path cost l -> s.
// ---------------------------------------------------------------------------
__global__ __launch_bounds__(256) void viterbi_chunk_mats(
    const float* __restrict__ llr, float* __restrict__ P, int nchunk) {
  const int wave = blockIdx.x * (blockDim.x >> 5) + (threadIdx.x >> 5);
  const int lane = threadIdx.x & 31;
  if (wave >= nchunk) return;

  float col[32];
#pragma unroll
  for (int s = 0; s < 32; ++s) col[s] = (s == lane) ? 0.0f : MPINF;

  const float2* lp2 = (const float2*)(llr + (size_t)wave * CHUNK * NSTATE);
  for (int t = 0; t < CHUNK; ++t) {
    float2 ll = lp2[(size_t)t * 32 + lane];          // llr[2*lane], llr[2*lane+1]
    __builtin_prefetch((const void*)(lp2 + (size_t)(t + 8) * 32 + lane), 0, 1);
    float nc[32];
#pragma unroll
    for (int s = 0; s < 32; ++s) {
      float la = __shfl(ll.x, s, 32);                // llr[2s]
      float lb = __shfl(ll.y, s, 32);                // llr[2s+1]
      nc[s] = fminf(col[(2 * s) & 31] + la, col[(2 * s + 1) & 31] + lb);
    }
#pragma unroll
    for (int s = 0; s < 32; ++s) col[s] = nc[s];
  }

  float* out = P + (size_t)wave * 1024;
#pragma unroll
  for (int s = 0; s < 32; ++s) out[s * 32 + lane] = col[s];
}

// ---------------------------------------------------------------------------
// Phase 2: single-wave serial chain of chunk boundary vectors.
// vb[c] = in_prob at the entry of chunk c;  v' [s] = min_j (P[s][j] + v[j]).
// ---------------------------------------------------------------------------
__global__ __launch_bounds__(32) void viterbi_boundaries(
    const float* __restrict__ P, float* __restrict__ vb, int nchunk) {
  const int lane = threadIdx.x & 31;
  float v = 0.0f;                                    // torch.zeros initial metric
  for (int c = 0; c < nchunk; ++c) {
    vb[c * 32 + lane] = v;
    const float4* row = (const float4*)(P + (size_t)c * 1024 + (size_t)lane * 32);
    __builtin_prefetch((const void*)(P + (size_t)(c + 1) * 1024 + (size_t)lane * 32), 0, 1);
    __builtin_prefetch((const void*)(P + (size_t)(c + 1) * 1024 + (size_t)lane * 32 + 16), 0, 1);
    float best = MPINF;
#pragma unroll
    for (int q = 0; q < 8; ++q) {
      float4 p4 = row[q];
      best = fminf(best, p4.x + __shfl(v, 4 * q + 0, 32));
      best = fminf(best, p4.y + __shfl(v, 4 * q + 1, 32));
      best = fminf(best, p4.z + __shfl(v, 4 * q + 2, 32));
      best = fminf(best, p4.w + __shfl(v, 4 * q + 3, 32));
    }
    v = best;
  }
}

// ---------------------------------------------------------------------------
// Phase 3: replay each chunk sequentially from its exact entry vector,
// emitting decoded bits (argmin BEFORE the update, as in the reference).
// ---------------------------------------------------------------------------
__global__ __launch_bounds__(256) void viterbi_decode(
    const float* __restrict__ llr, const float* __restrict__ vb,
    float* __restrict__ out, int nchunk) {
  const int wave = blockIdx.x * (blockDim.x >> 5) + (threadIdx.x >> 5);
  const int lane = threadIdx.x & 31;
  if (wave >= nchunk) return;

  float v = vb[wave * 32 + lane];
  const float2* lp2 = (const float2*)(llr + (size_t)wave * CHUNK * NSTATE);
  float* op = out + (size_t)wave * CHUNK;

  for (int t = 0; t < CHUNK; ++t) {
    int midx = wave_argmin32(v, lane);               // first-occurrence argmin < 32
    if (lane == 0) op[t] = (float)(midx & 1);
    float2 ll = lp2[(size_t)t * 32 + lane];
    __builtin_prefetch((const void*)(lp2 + (size_t)(t + 8) * 32 + lane), 0, 1);
    float p0 = __shfl(v, (2 * lane) & 31, 32);
    float p1 = __shfl(v, (2 * lane + 1) & 31, 32);
    v = fminf(p0 + ll.x, p1 + ll.y);
  }
}

// ---------------------------------------------------------------------------
// Fallback: fully fused single-wave sequential detector (used if T % CHUNK != 0 or the
// workspace is too small to hold llrs + chunk matrices). Exact semantics,
// llrs computed on the fly (lane l owns states 2l and 2l+1 of the llr row).
// ---------------------------------------------------------------------------
__global__ __launch_bounds__(32) void viterbi_fused_seq(
    const float* __restrict__ rx, const float* __restrict__ w1,
    const float* __restrict__ b1, const float* __restrict__ w2,
    const float* __restrict__ b2, float* __restrict__ out, int T) {
  const int lane = threadIdx.x & 31;
  const float* w2a = w2 + (size_t)(2 * lane) * HIDDEN;
  const float* w2b = w2 + (size_t)(2 * lane + 1) * HIDDEN;
  const float bba = b2[2 * lane], bbb = b2[2 * lane + 1];
  float v = 0.0f;
  for (int t = 0; t < T; ++t) {
    int midx = wave_argmin32(v, lane);
    if (lane == 0) out[t] = (float)(midx & 1);
    const float x = rx[t];
    float a0 = 0.f, a1 = 0.f;
#pragma unroll 8
    for (int h = 0; h < HIDDEN; ++h) {
      float hv = fmaxf(fmaf(x, w1[h], b1[h]), 0.0f);
      a0 = fmaf(hv, w2a[h], a0);
      a1 = fmaf(hv, w2b[h], a1);
    }
    float l0 = -(a0 + bba);                          // llr[2*lane]
    float l1 = -(a1 + bbb);                          // llr[2*lane+1]
    float p0 = __shfl(v, (2 * lane) & 31, 32);
    float p1 = __shfl(v, (2 * lane + 1) & 31, 32);
    v = fminf(p0 + l0, p1 + l1);
  }
}

// ---------------------------------------------------------------------------
extern "C" void kernel_launch(void* const* d_in, const int* in_sizes, int n_in,
                              void* d_out, int out_size, void* d_ws, size_t ws_size,
                              hipStream_t stream) {
  const float* rx = (const float*)d_in[0];
  const float* w1 = (const float*)d_in[1];
  const float* b1 = (const float*)d_in[2];
  const float* w2 = (const float*)d_in[3];
  const float* b2 = (const float*)d_in[4];
  float* out = (float*)d_out;
  const int T = in_sizes[0];

  const size_t llrBytes = (size_t)T * NSTATE * sizeof(float);
  const int    nchunk   = T / CHUNK;
  const size_t pBytes   = (size_t)nchunk * 32 * 32 * sizeof(float);
  const size_t vbBytes  = (size_t)nchunk * 32 * sizeof(float);

  const bool fast = (T % CHUNK == 0) && (T % 16 == 0) &&
                    (ws_size >= llrBytes + pBytes + vbBytes);

  if (!fast) {
    viterbi_fused_seq<<<1, 32, 0, stream>>>(rx, w1, b1, w2, b2, out, T);
    return;
  }

  float* llr = (float*)d_ws;
  float* P   = (float*)((char*)d_ws + llrBytes);
  float* vb  = (float*)((char*)d_ws + llrBytes + pBytes);

  const int mlpWaves = T >> 4;                       // 16 timesteps per wave
  mlp_llr_wmma<<<(mlpWaves + 7) / 8, 256, 0, stream>>>(rx, w1, b1, w2, b2, llr, T);
  viterbi_chunk_mats<<<(nchunk + 7) / 8, 256, 0, stream>>>(llr, P, nchunk);
  viterbi_boundaries<<<1, 32, 0, stream>>>(P, vb, nchunk);
  viterbi_decode<<<(nchunk + 7) / 8, 256, 0, stream>>>(llr, vb, out, nchunk);
}